// TimeReasoning_76270029242483
// MI455X (gfx1250) — compile-verified
//
#include <hip/hip_runtime.h>
#include <math.h>

// MI455X / gfx1250, wave32. f32 WMMA (16x16x4) for both GEMMs; edge GEMM fused
// with the per-step rel projection so tm (67 MB) never hits HBM.

typedef __attribute__((ext_vector_type(2))) float v2f;
typedef __attribute__((ext_vector_type(8))) float v8f;

// Only check in the DEVICE pass; this hipcc defines __AMDGCN__ even for host.
#if defined(__HIP_DEVICE_COMPILE__) && !__has_builtin(__builtin_amdgcn_wmma_f32_16x16x4_f32)
#error "missing __builtin_amdgcn_wmma_f32_16x16x4_f32 on gfx1250"
#endif

#define NEG_BIG  (-100000.0f)
#define TINYF    (1.17549435e-38f)

__device__ __forceinline__ unsigned rotl32(unsigned x, int r) {
    return (x << r) | (x >> (32 - r));
}

// JAX threefry2x32 (20 rounds, rotation schedule {13,15,26,6}/{17,29,16,24})
__device__ void threefry2x32(unsigned k0, unsigned k1, unsigned x0, unsigned x1,
                             unsigned* o0, unsigned* o1) {
    unsigned ks2 = k0 ^ k1 ^ 0x1BD11BDAu;
    x0 += k0; x1 += k1;
#define TF_R(r) { x0 += x1; x1 = rotl32(x1, (r)); x1 ^= x0; }
#define TF_G(a,b,c,d) TF_R(a) TF_R(b) TF_R(c) TF_R(d)
    TF_G(13,15,26,6)  x0 += k1;  x1 += ks2 + 1u;
    TF_G(17,29,16,24) x0 += ks2; x1 += k0  + 2u;
    TF_G(13,15,26,6)  x0 += k0;  x1 += k1  + 3u;
    TF_G(17,29,16,24) x0 += k1;  x1 += ks2 + 4u;
    TF_G(13,15,26,6)  x0 += ks2; x1 += k0  + 5u;
#undef TF_G
#undef TF_R
    *o0 = x0; *o1 = x1;
}

// ---------------------------------------------------------------- K0: init
__global__ void tr_init_kernel(float* out, float* racc) {
    int t = blockIdx.x * blockDim.x + threadIdx.x;
    if (t == 0) out[0] = 0.f;
    if (t < 128) racc[t] = 0.f;
}

// ---------------------------------------------------------------- K1: prep
// One block per (i, b), i in [0,8), b in [0,16). Computes:
//   wvec[(i*16+b)*256+k] = relu(q_emb[i,b]·W_q[k] + b_q[k]) * W_o[k]
//   post[(b*8+i)*64+n], z (gumbel softmax sample), termpost = Σ z·logsoftmax(post)
//   h_all[(i*16+b)*1024+h] = Σ_n obj[b,n,h]·softmax(post)[n]
__global__ __launch_bounds__(256)
void tr_prep_kernel(const float* __restrict__ q_emb, const float* __restrict__ obj,
                    const float* __restrict__ logits, const float* __restrict__ v_mask,
                    const float* __restrict__ W_q, const float* __restrict__ b_q,
                    const float* __restrict__ W_o,
                    float* wvec, float* post, float* zbuf, float* termpost, float* h_all) {
    int i = blockIdx.x >> 4;
    int b = blockIdx.x & 15;
    int tid = threadIdx.x;
    __shared__ float sm_post[64], sm_act[64], sm_z[64], sm_red[1];

    { // qv / wvec (k = tid, 256 outputs)
        const float* q = q_emb + (size_t)(i * 16 + b) * 512;
        const float* w = W_q + (size_t)tid * 512;
        float s = b_q[tid];
        for (int m = 0; m < 512; m++) s += q[m] * w[m];
        s = fmaxf(s, 0.f);
        wvec[(size_t)(i * 16 + b) * 256 + tid] = s * W_o[tid];
    }

    if (tid < 64) { // post logits (n = tid)
        const float* lg = logits + (size_t)((b * 9 + i) * 64 + tid) * 6;
        float s = 0.f;
        for (int c = 0; c < 6; c++) {
            float x = lg[c];
            if (x == -INFINITY) x = 0.f;
            s += x;
        }
        bool vbad = (v_mask[b * 64 + tid] == 0.f);
        float p = vbad ? NEG_BIG : s;
        sm_post[tid] = p;
        post[(size_t)(b * 8 + i) * 64 + tid] = p;
    }
    __syncthreads();

    if (tid < 64) { // gumbel via threefry: key = fold_in(key(42), i)
        unsigned fk0, fk1;
        threefry2x32(0u, 42u, 0u, (unsigned)i, &fk0, &fk1);
        unsigned j = (unsigned)(b * 64 + tid);          // element in (16,64)
        unsigned p0 = (j < 512u) ? j : (j - 512u);      // counter pairing
        unsigned y0, y1;
        threefry2x32(fk0, fk1, p0, p0 + 512u, &y0, &y1);
        unsigned bits = (j < 512u) ? y0 : y1;
        float u = __uint_as_float((bits >> 9) | 0x3F800000u) - 1.0f;
        u = fmaxf(u * (1.0f - TINYF) + TINYF, TINYF);
        float g = -logf(-logf(u));
        sm_z[tid] = sm_post[tid] + g;
        bool vbad = (v_mask[b * 64 + tid] == 0.f);
        sm_act[tid] = vbad ? -INFINITY : sm_post[tid];
    }
    __syncthreads();
    if (tid == 0) { // small serial softmaxes over 64
        float mx = -INFINITY; for (int n = 0; n < 64; n++) mx = fmaxf(mx, sm_act[n]);
        float den = 0.f;      for (int n = 0; n < 64; n++) den += expf(sm_act[n] - mx);
        for (int n = 0; n < 64; n++) sm_act[n] = expf(sm_act[n] - mx) / den;
        float mz = -INFINITY; for (int n = 0; n < 64; n++) mz = fmaxf(mz, sm_z[n]);
        float dz = 0.f;       for (int n = 0; n < 64; n++) dz += expf(sm_z[n] - mz);
        float mp = -INFINITY; for (int n = 0; n < 64; n++) mp = fmaxf(mp, sm_post[n]);
        float dp = 0.f;       for (int n = 0; n < 64; n++) dp += expf(sm_post[n] - mp);
        float lsep = mp + logf(dp);
        float term = 0.f;
        for (int n = 0; n < 64; n++) {
            float z = expf(sm_z[n] - mz) / dz;
            sm_z[n] = z;
            term += z * (sm_post[n] - lsep);
        }
        termpost[b * 8 + i] = term;
        sm_red[0] = 0.f;
    }
    __syncthreads();
    if (tid < 64) zbuf[(size_t)(b * 8 + i) * 64 + tid] = sm_z[tid];

    for (int h = tid; h < 1024; h += 256) { // h_act
        float s = 0.f;
        const float* ob = obj + (size_t)b * 64 * 1024 + h;
        for (int n = 0; n < 64; n++) s += ob[(size_t)n * 1024] * sm_act[n];
        h_all[(size_t)(i * 16 + b) * 1024 + h] = s;
    }
}

// ------------------------------------------------- K2: fused edge GEMM + rel
// rows = (b,n,m) flattened (65536 rows, 256 K). Each wave: one 16-row tile.
// tm_row = relu(edge_row·W_r.T + b_r) computed in f32 WMMA tiles; immediately
// contracted against the 8 step weight vectors; rel written, tm discarded.
__global__ __launch_bounds__(256)
void tr_edge_rel_kernel(const float* __restrict__ edge, const float* __restrict__ W_r,
                        const float* __restrict__ b_r, const float* __restrict__ b_o,
                        const float* __restrict__ wvec, float* __restrict__ rel) {
    int lane = threadIdx.x & 31;
    int wave = threadIdx.x >> 5;
    int rowTile = blockIdx.x * 8 + wave;   // 0..4095
    int row0 = rowTile * 16;               // 65536 total rows
    int b = row0 >> 12;                    // 4096 rows per batch
    int lr = lane & 15;
    int hi = lane >> 4;                    // K half select per WMMA layout
    const float* Abase = edge + (size_t)(row0 + lr) * 256 + 2 * hi;

    float relp[64];
#pragma unroll
    for (int x = 0; x < 64; x++) relp[x] = 0.f;

    for (int o = 0; o < 16; o++) {         // 16 output-channel tiles of W_r
        const float* Bbase = W_r + (size_t)(o * 16 + lr) * 256 + 2 * hi;
        v8f c = {0.f, 0.f, 0.f, 0.f, 0.f, 0.f, 0.f, 0.f};
#pragma unroll 8
        for (int k = 0; k < 64; k++) {     // K = 256 in steps of 4
            v2f a  = *(const v2f*)(Abase + 4 * k);
            v2f bb = *(const v2f*)(Bbase + 4 * k);
            c = __builtin_amdgcn_wmma_f32_16x16x4_f32(false, a, false, bb,
                                                      (short)0, c, false, false);
        }
        float bias = b_r[o * 16 + lr];
        float wv[8];
#pragma unroll
        for (int ii = 0; ii < 8; ii++)
            wv[ii] = wvec[(size_t)(ii * 16 + b) * 256 + o * 16 + lr];
#pragma unroll
        for (int jj = 0; jj < 8; jj++) {
            float t = fmaxf(c[jj] + bias, 0.f);   // tm element (relu + b_r)
#pragma unroll
            for (int ii = 0; ii < 8; ii++) relp[ii * 8 + jj] += t * wv[ii];
        }
    }
    // reduce the channel partials across the 16 lanes of each half-wave
#pragma unroll
    for (int msk = 1; msk < 16; msk <<= 1)
#pragma unroll
        for (int x = 0; x < 64; x++)
            relp[x] += __shfl_xor(relp[x], msk, 32);

    if (lr == 0) {
        float bo = b_o[0];
        int rib0 = row0 - b * 4096;        // row within batch (n*64 + m)
#pragma unroll
        for (int jj = 0; jj < 8; jj++) {
            int rib = rib0 + jj + 8 * hi;
            int n = rib >> 6, m = rib & 63;
#pragma unroll
            for (int ii = 0; ii < 8; ii++) {
                float r = fmaxf(relp[ii * 8 + jj] + bo, 0.f);  // fc relu + b_o
                rel[((size_t)(b * 8 + ii) * 64 + n) * 64 + m] = r;
            }
        }
    }
}

// -------------------------------------------- K3: recon GEMM + BCE epilogue
// (128 x 1024) h_all  x  W_dec.T (1024 x 2048); BCE-with-logits vs e_label
// reduced over T into recon_acc[b*8+i].
__global__ __launch_bounds__(256)
void tr_recon_kernel(const float* __restrict__ h_all, const float* __restrict__ W_dec,
                     const float* __restrict__ b_dec, const float* __restrict__ e_label,
                     float* __restrict__ recon_acc) {
    int lane = threadIdx.x & 31;
    int wave = threadIdx.x >> 5;
    int tile = blockIdx.x * 8 + wave;      // 0..1023 = 8 row tiles x 128 col tiles
    int rt = tile >> 7;
    int ot = tile & 127;
    int lr = lane & 15, hi = lane >> 4;
    int r0 = rt * 16, t0 = ot * 16;
    const float* Abase = h_all + (size_t)(r0 + lr) * 1024 + 2 * hi;
    const float* Bbase = W_dec + (size_t)(t0 + lr) * 1024 + 2 * hi;
    v8f c = {0.f, 0.f, 0.f, 0.f, 0.f, 0.f, 0.f, 0.f};
#pragma unroll 8
    for (int k = 0; k < 256; k++) {        // K = 1024 in steps of 4
        v2f a  = *(const v2f*)(Abase + 4 * k);
        v2f bb = *(const v2f*)(Bbase + 4 * k);
        c = __builtin_amdgcn_wmma_f32_16x16x4_f32(false, a, false, bb,
                                                  (short)0, c, false, false);
    }
    int t = t0 + lr;
    float bias = b_dec[t];
    float acc[8];
#pragma unroll
    for (int jj = 0; jj < 8; jj++) {
        int r = r0 + jj + 8 * hi;          // row = i*16 + b
        int i = r >> 4, b = r & 15;
        float x = c[jj] + bias;
        float y = e_label[(size_t)(b * 9 + i) * 2048 + t];
        acc[jj] = fmaxf(x, 0.f) - x * y + log1pf(expf(-fabsf(x)));
    }
#pragma unroll
    for (int msk = 1; msk < 16; msk <<= 1)
#pragma unroll
        for (int jj = 0; jj < 8; jj++) acc[jj] += __shfl_xor(acc[jj], msk, 32);
    if (lr == 0) {
#pragma unroll
        for (int jj = 0; jj < 8; jj++) {
            int r = r0 + jj + 8 * hi;
            int i = r >> 4, b = r & 15;
            atomicAdd(recon_acc + (b * 8 + i), acc[jj] * (1.0f / 2048.0f));
        }
    }
}

// ------------------------------------------------------------- K4: finalize
// One block per (b,i): relation softmax over n (axis=1), prior, l_forward,
// loss accumulation into scalar out.
__global__ __launch_bounds__(64)
void tr_final_kernel(const float* __restrict__ rel, const float* __restrict__ r_mask,
                     const float* __restrict__ post, const float* __restrict__ zbuf,
                     const float* __restrict__ termpost, const float* __restrict__ recon_acc,
                     const float* __restrict__ e_mask, float* out) {
    int b = blockIdx.x >> 3;
    int i = blockIdx.x & 7;
    int tid = threadIdx.x;                 // 64
    __shared__ float sm_mx[64], sm_w[64], sm_prior[64], sm_red[1];
    float lf = 0.f;
    if (i > 0) {
        int ip = i - 1;
        const float* R = rel + (size_t)(b * 8 + ip) * 64 * 64;
        const float* M = r_mask + (size_t)b * 64 * 64;
        { // per-column (m) softmax stats over n
            int m = tid;
            float mx = -INFINITY;
            for (int n = 0; n < 64; n++)
                if (M[n * 64 + m] != 0.f) mx = fmaxf(mx, R[n * 64 + m]);
            float den = 0.f;
            if (mx > -INFINITY)
                for (int n = 0; n < 64; n++)
                    if (M[n * 64 + m] != 0.f) den += expf(R[n * 64 + m] - mx);
            sm_mx[m] = mx;
            sm_w[m] = (den > 0.f) ? post[(size_t)(b * 8 + ip) * 64 + m] / den : 0.f;
        }
        __syncthreads();
        { // prior[n] = Σ_m relmat[n,m] * post_{i-1}[m]
            int n = tid;
            float s = 0.f;
            for (int m = 0; m < 64; m++)
                if (M[n * 64 + m] != 0.f)
                    s += expf(R[n * 64 + m] - sm_mx[m]) * sm_w[m];
            sm_prior[n] = s;
        }
        __syncthreads();
        if (tid == 0) {
            float mp = -INFINITY; for (int n = 0; n < 64; n++) mp = fmaxf(mp, sm_prior[n]);
            float dp = 0.f;       for (int n = 0; n < 64; n++) dp += expf(sm_prior[n] - mp);
            float lse = mp + logf(dp);
            float lp = 0.f;
            for (int n = 0; n < 64; n++)
                lp += zbuf[(size_t)(b * 8 + i) * 64 + n] * (sm_prior[n] - lse);
            sm_red[0] = termpost[b * 8 + i] - lp;  // l_forward
        }
        __syncthreads();
        lf = sm_red[0];
    }
    if (tid == 0) {
        float loss = recon_acc[b * 8 + i] + ((i > 0) ? lf : 0.f);
        loss *= e_mask[b * 9 + i];
        atomicAdd(out, loss * (1.0f / 16.0f));
    }
}

// ---------------------------------------------------------------- launcher
extern "C" void kernel_launch(void* const* d_in, const int* in_sizes, int n_in,
                              void* d_out, int out_size, void* d_ws, size_t ws_size,
                              hipStream_t stream) {
    const float* q_emb   = (const float*)d_in[0];   // (9,16,512)
    const float* obj     = (const float*)d_in[1];   // (16,64,1024)
    const float* e_label = (const float*)d_in[2];   // (16,9,2048)
    // d_in[3], d_in[4] (e_s, e_e) unused by the reference
    const float* e_mask  = (const float*)d_in[5];   // (16,9)
    const float* edge    = (const float*)d_in[6];   // (16,64,64,256)
    const float* r_mask  = (const float*)d_in[7];   // (16,64,64)
    const float* logits  = (const float*)d_in[8];   // (16,9,64,6)
    const float* v_mask  = (const float*)d_in[9];   // (16,64)
    const float* W_r     = (const float*)d_in[10];  // (256,256)
    const float* b_r     = (const float*)d_in[11];  // (256)
    const float* W_q     = (const float*)d_in[12];  // (256,512)
    const float* b_q     = (const float*)d_in[13];  // (256)
    const float* W_o     = (const float*)d_in[14];  // (1,256)
    const float* b_o     = (const float*)d_in[15];  // (1)
    const float* W_dec   = (const float*)d_in[16];  // (2048,1024)
    const float* b_dec   = (const float*)d_in[17];  // (2048)
    float* out = (float*)d_out;

    float* ws = (float*)d_ws;               // ~2.82 MB of f32 scratch used
    float* wvec      = ws;                  // 8*16*256   = 32768
    float* post      = wvec + 32768;        // 16*8*64    = 8192
    float* zbuf      = post + 8192;         // 16*8*64    = 8192
    float* termpost  = zbuf + 8192;         // 128
    float* h_all     = termpost + 128;      // 128*1024   = 131072
    float* rel       = h_all + 131072;      // 16*8*64*64 = 524288
    float* recon_acc = rel + 524288;        // 128

    tr_init_kernel<<<1, 256, 0, stream>>>(out, recon_acc);
    tr_prep_kernel<<<128, 256, 0, stream>>>(q_emb, obj, logits, v_mask, W_q, b_q, W_o,
                                            wvec, post, zbuf, termpost, h_all);
    tr_edge_rel_kernel<<<512, 256, 0, stream>>>(edge, W_r, b_r, b_o, wvec, rel);
    tr_recon_kernel<<<128, 256, 0, stream>>>(h_all, W_dec, b_dec, e_label, recon_acc);
    tr_final_kernel<<<128, 64, 0, stream>>>(rel, r_mask, post, zbuf, termpost,
                                            recon_acc, e_mask, out);
}